// GNNEncoder_4157528342735
// MI455X (gfx1250) — compile-verified
//
#include <hip/hip_runtime.h>
#include <hip/hip_bf16.h>
#include <stdint.h>
#include <stddef.h>

typedef __attribute__((ext_vector_type(2))) float v2f;
typedef __attribute__((ext_vector_type(8))) float v8f;

#define NEG_SLOPE 0.2f
#define GAT_EPS 1e-16f

// ---------- order-preserving float<->uint map for atomic segment-max ----------
__device__ __forceinline__ unsigned f32_ord(float f) {
    unsigned u = __float_as_uint(f);
    return (u & 0x80000000u) ? ~u : (u | 0x80000000u);
}
__device__ __forceinline__ float f32_unord(unsigned u) {
    unsigned b = (u & 0x80000000u) ? (u & 0x7FFFFFFFu) : ~u;
    return __uint_as_float(b);
}
#define ORD_NEG_INF 0x007FFFFFu   // f32_ord(-inf)

// ---------- init kernels ----------
__global__ void fill_f32(float* __restrict__ p, float v, int n) {
    int i = blockIdx.x * blockDim.x + threadIdx.x;
    if (i < n) p[i] = v;
}
__global__ void fill_u32(unsigned* __restrict__ p, unsigned v, int n) {
    int i = blockIdx.x * blockDim.x + threadIdx.x;
    if (i < n) p[i] = v;
}

// ---------- f32 WMMA GEMM with LDS staging ----------
// D[M,Nc] = A[M,K] @ B[K,Nc].  K <= 128, Nc == 64 (multiple of 16).
// Block = 256 threads = 8 waves; block covers 32 rows x Nc cols
// (2 row-tiles x 4 col-tiles, one 16x16 tile per wave).
// A-slab (32 x K) and all of B (K x Nc) staged in LDS, padded to avoid
// bank conflicts, then the K-loop feeds V_WMMA_F32_16X16X4_F32 from LDS.
//
// WMMA f32 16x16x4 register layouts (ISA 7.12.2):
//  A 16x4 : lanes 0-15 row M=lane  k={k0,k0+1}; lanes 16-31 row M=lane-16 k={k0+2,k0+3}
//  B 4x16 : lanes 0-15 col N=lane  k={k0,k0+1}; lanes 16-31 col N=lane-16 k={k0+2,k0+3}
//  C/D    : VGPR r -> M=r (lanes 0-15) / M=r+8 (lanes 16-31), N = lane&15
#define GEMM_KMAX  128
#define GEMM_NMAX  64
#define BLK_ROWS   32
#define A_STRIDE   (GEMM_KMAX + 2)   // bank stride 2 between consecutive rows' columns
#define B_STRIDE   (GEMM_NMAX + 16)  // rows k and k+2 land 32 banks apart

__global__ __launch_bounds__(256) void gemm_wmma_f32_lds(
        const float* __restrict__ A, const float* __restrict__ B,
        float* __restrict__ D, int M, int K, int Nc) {
    __shared__ float As[BLK_ROWS * A_STRIDE];
    __shared__ float Bs[GEMM_KMAX * B_STRIDE];

    const int tid     = threadIdx.x;
    const int lane    = tid & 31;
    const int wave    = tid >> 5;
    const int tilesN  = Nc >> 4;                 // 4
    const int rowBase = blockIdx.x * BLK_ROWS;

    // ---- stage B (entire K x Nc weight matrix) ----
    for (int i = tid; i < K * Nc; i += 256) {
        int r = i / Nc, c = i - r * Nc;
        Bs[r * B_STRIDE + c] = B[i];
    }
    // ---- stage A slab (32 rows x K), zero-fill out-of-range rows ----
    for (int i = tid; i < BLK_ROWS * K; i += 256) {
        int r = i / K, c = i - r * K;
        int gr = rowBase + r;
        As[r * A_STRIDE + c] = (gr < M) ? A[(size_t)gr * K + c] : 0.f;
    }
    __syncthreads();

    const int rowTile = wave / tilesN;           // 0..1
    const int colTile = wave - rowTile * tilesN; // 0..3
    if (rowBase + rowTile * 16 >= M) return;     // wave-uniform guard

    const int halfSel = lane >> 4;
    const int l15     = lane & 15;

    const float* aptr = As + (rowTile * 16 + l15) * A_STRIDE;
    const float* bptr = Bs + colTile * 16 + l15;

    v8f acc = {};
    #pragma unroll 4
    for (int k0 = 0; k0 < K; k0 += 4) {
        const int ka = k0 + halfSel * 2;
        v2f a, b;
        a.x = aptr[ka + 0];
        a.y = aptr[ka + 1];
        b.x = bptr[(ka + 0) * B_STRIDE];
        b.y = bptr[(ka + 1) * B_STRIDE];
        acc = __builtin_amdgcn_wmma_f32_16x16x4_f32(
            false, a, false, b, (short)0, acc, false, false);
    }

    float* drow = D + (size_t)(rowBase + rowTile * 16 + halfSel * 8) * Nc
                    + (colTile * 16 + l15);
    #pragma unroll
    for (int r = 0; r < 8; ++r)
        drow[(size_t)r * Nc] = acc[r];
}

// ---------- per-node attention logits: as[n]=h[n]·a_src, ad[n]=h[n]·a_dst ----------
__global__ void node_alpha(const float* __restrict__ h, const float* __restrict__ a_src,
                           const float* __restrict__ a_dst, float* __restrict__ as,
                           float* __restrict__ ad, int N, int C) {
    int n = blockIdx.x * blockDim.x + threadIdx.x;
    if (n >= N) return;
    const float* row = h + (size_t)n * (size_t)C;
    float s = 0.f, d = 0.f;
    for (int c = 0; c < C; ++c) {
        float v = row[c];
        s = fmaf(v, a_src[c], s);
        d = fmaf(v, a_dst[c], d);
    }
    as[n] = s;
    ad[n] = d;
}

// ---------- edge pass A: e = LeakyReLU(as[src]+ad[dst]); segment max into mord ----------
__global__ void edge_max(const long long* __restrict__ src, const long long* __restrict__ dst,
                         const float* __restrict__ as, const float* __restrict__ ad,
                         float* __restrict__ eout, unsigned* __restrict__ mord, int E) {
    int e = blockIdx.x * blockDim.x + threadIdx.x;
    if (e >= E) return;
    int s = (int)src[e], d = (int)dst[e];
    float v = as[s] + ad[d];
    v = (v > 0.f) ? v : NEG_SLOPE * v;
    eout[e] = v;
    atomicMax(&mord[d], f32_ord(v));
}

// ---------- edge pass B: p = exp(e - m[dst]); segment sum into ssum ----------
__global__ void edge_expsum(const long long* __restrict__ dst, float* __restrict__ ep,
                            const unsigned* __restrict__ mord, float* __restrict__ ssum, int E) {
    int e = blockIdx.x * blockDim.x + threadIdx.x;
    if (e >= E) return;
    int d = (int)dst[e];
    float m = f32_unord(mord[d]);
    float p = expf(ep[e] - m);
    ep[e] = p;                    // overwrite e with p
    atomicAdd(&ssum[d], p);
}

// ---------- edge pass C: acc[dst] += (p/(s[dst]+eps)) * h[src]  (one wave per edge) ----------
__global__ void edge_scatter(const long long* __restrict__ src, const long long* __restrict__ dst,
                             const float* __restrict__ p, const float* __restrict__ ssum,
                             const float* __restrict__ h, float* __restrict__ acc, int E, int C) {
    int wave = blockIdx.x * (blockDim.x >> 5) + (threadIdx.x >> 5);
    int lane = threadIdx.x & 31;
    if (wave >= E) return;
    int s = (int)src[wave], d = (int)dst[wave];
    float alpha = p[wave] / (ssum[d] + GAT_EPS);
    const float* hs = h + (size_t)s * (size_t)C;
    float* od = acc + (size_t)d * (size_t)C;
    for (int c = lane; c < C; c += 32)
        atomicAdd(&od[c], alpha * hs[c]);
}

// ---------- epilogue: out = acc + b  (optional ReLU) ----------
__global__ void bias_act(const float* __restrict__ acc, const float* __restrict__ b,
                         float* __restrict__ out, int total, int C, int doRelu) {
    int i = blockIdx.x * blockDim.x + threadIdx.x;
    if (i >= total) return;
    float v = acc[i] + b[i % C];
    out[i] = doRelu ? fmaxf(v, 0.f) : v;
}

// =======================================================================
extern "C" void kernel_launch(void* const* d_in, const int* in_sizes, int n_in,
                              void* d_out, int out_size, void* d_ws, size_t ws_size,
                              hipStream_t stream) {
    const float*     x      = (const float*)d_in[0];
    const long long* ei     = (const long long*)d_in[1];
    const float*     W1     = (const float*)d_in[2];
    const float*     a_src1 = (const float*)d_in[3];
    const float*     a_dst1 = (const float*)d_in[4];
    const float*     b1     = (const float*)d_in[5];
    const float*     W2     = (const float*)d_in[6];
    const float*     a_src2 = (const float*)d_in[7];
    const float*     a_dst2 = (const float*)d_in[8];
    const float*     b2     = (const float*)d_in[9];

    const int H   = in_sizes[3];            // 64
    const int Fin = in_sizes[2] / H;        // 128
    const int N   = in_sizes[0] / Fin;      // 50000
    const int E   = in_sizes[1] / 2;        // 800000
    const int C   = in_sizes[7];            // 64

    const long long* src = ei;
    const long long* dst = ei + E;

    // ---- workspace layout (with reuse) ----
    const int F = (H > C) ? H : C;
    char*  ws   = (char*)d_ws;
    size_t off  = 0;
    float* bufA = (float*)(ws + off); off += (size_t)N * F * 4;  // h1, later g2 (=h2@W2)
    float* bufB = (float*)(ws + off); off += (size_t)N * F * 4;  // acc (both layers)
    float* bufC = (float*)(ws + off); off += (size_t)N * F * 4;  // h2 = relu(acc1+b1)
    float* as   = (float*)(ws + off); off += (size_t)N * 4;
    float* ad   = (float*)(ws + off); off += (size_t)N * 4;
    unsigned* mord = (unsigned*)(ws + off); off += (size_t)N * 4;
    float* ssum = (float*)(ws + off); off += (size_t)N * 4;
    float* ep   = (float*)(ws + off); off += (size_t)E * 4;      // edge e, then p

    const int BT = 256;                       // threads per block
    const int wavesPB = BT / 32;
    auto cdiv = [](long long a, long long b) { return (int)((a + b - 1) / b); };

    // ================= Layer 1 =================
    {
        gemm_wmma_f32_lds<<<cdiv(N, BLK_ROWS), BT, 0, stream>>>(x, W1, bufA, N, Fin, H);

        node_alpha<<<cdiv(N, BT), BT, 0, stream>>>(bufA, a_src1, a_dst1, as, ad, N, H);

        fill_u32<<<cdiv(N, BT), BT, 0, stream>>>(mord, ORD_NEG_INF, N);
        fill_f32<<<cdiv(N, BT), BT, 0, stream>>>(ssum, 0.f, N);
        fill_f32<<<cdiv((long long)N * H, BT), BT, 0, stream>>>(bufB, 0.f, N * H);

        edge_max   <<<cdiv(E, BT), BT, 0, stream>>>(src, dst, as, ad, ep, mord, E);
        edge_expsum<<<cdiv(E, BT), BT, 0, stream>>>(dst, ep, mord, ssum, E);
        edge_scatter<<<cdiv(E, wavesPB), BT, 0, stream>>>(src, dst, ep, ssum, bufA, bufB, E, H);

        bias_act<<<cdiv((long long)N * H, BT), BT, 0, stream>>>(bufB, b1, bufC, N * H, H, 1);
    }

    // ================= Layer 2 =================
    {
        gemm_wmma_f32_lds<<<cdiv(N, BLK_ROWS), BT, 0, stream>>>(bufC, W2, bufA, N, H, C);

        node_alpha<<<cdiv(N, BT), BT, 0, stream>>>(bufA, a_src2, a_dst2, as, ad, N, C);

        fill_u32<<<cdiv(N, BT), BT, 0, stream>>>(mord, ORD_NEG_INF, N);
        fill_f32<<<cdiv(N, BT), BT, 0, stream>>>(ssum, 0.f, N);
        fill_f32<<<cdiv((long long)N * C, BT), BT, 0, stream>>>(bufB, 0.f, N * C);

        edge_max   <<<cdiv(E, BT), BT, 0, stream>>>(src, dst, as, ad, ep, mord, E);
        edge_expsum<<<cdiv(E, BT), BT, 0, stream>>>(dst, ep, mord, ssum, E);
        edge_scatter<<<cdiv(E, wavesPB), BT, 0, stream>>>(src, dst, ep, ssum, bufA, bufB, E, C);

        bias_act<<<cdiv((long long)N * C, BT), BT, 0, stream>>>(bufB, b2, (float*)d_out, N * C, C, 0);
    }
}